// GaussianMixtureUnif_88012469830601
// MI455X (gfx1250) — compile-verified
//
#include <hip/hip_runtime.h>

// CDNA5 / gfx1250, wave32. Score GEMM via V_WMMA_F32_16X16X4_F32.
typedef __attribute__((ext_vector_type(2))) float v2f;
typedef __attribute__((ext_vector_type(8))) float v8f;

#define N_PTS   131072
#define NC      512
#define ROWS_WG 128
#define THREADS 256
#define CTILES  (NC / 16)

// workspace layout (float offsets); total ~551KB
#define WS_MUS   0        // float4[512]: (mu*inv_s2, b_c)
#define WS_MU2   2048     // float[512]:  ||mu||^2
#define WS_S     2560     // float[512]:  sum_n gamma
#define WS_TX    3072     // float[512]
#define WS_TY    3584     // float[512]
#define WS_TZ    4096     // float[512]
#define WS_GD2   4608     // float[1]
#define WS_REC2  4624     // float4[512]: (mu_new, A_c)   (16B aligned)
#define WS_SCAL  6672     // [0]=1/(2*sn^2)  [1]=3*log(sn)
#define WS_L     6680     // float[N]: per-row logsumexp

__device__ __forceinline__ v8f wmma_score(v2f A, v2f B, v8f Cm) {
  // D = A(16x4 f32) x B(4x16 f32) + C(16x16 f32)
  return __builtin_amdgcn_wmma_f32_16x16x4_f32(false, A, false, B, (short)0, Cm,
                                               false, false);
}

// ---------------- K0: per-component tables + zero accumulators ----------------
__global__ __launch_bounds__(512) void k0_prep(const float* __restrict__ mu,
                                               const float* __restrict__ w,
                                               const float* __restrict__ sigma,
                                               float* __restrict__ ws) {
  int c = threadIdx.x;
  float s = sigma[0];
  float invs2 = 1.0f / (s * s);
  if (c < NC) {
    float mx = mu[c * 3 + 0], my = mu[c * 3 + 1], mz = mu[c * 3 + 2];
    float m2 = mx * mx + my * my + mz * mz;
    ((float4*)(ws + WS_MUS))[c] =
        make_float4(mx * invs2, my * invs2, mz * invs2, w[c] - 0.5f * m2 * invs2);
    ws[WS_MU2 + c] = m2;
    ws[WS_S + c] = 0.0f;
    ws[WS_TX + c] = 0.0f;
    ws[WS_TY + c] = 0.0f;
    ws[WS_TZ + c] = 0.0f;
  }
  if (c == 0) ws[WS_GD2] = 0.0f;
}

// ---------------- K1: E-step (row softmax) + column accumulators ----------------
__global__ __launch_bounds__(THREADS) void k1_estep(const float* __restrict__ X,
                                                    const float* __restrict__ sigma,
                                                    float* __restrict__ ws) {
  __shared__ float Xs[ROWS_WG * 3];
  __shared__ float Sacc[NC], Txa[NC], Tya[NC], Tza[NC];
  const int tid = threadIdx.x;
  const int gRow0 = blockIdx.x * ROWS_WG;
  for (int i = tid; i < ROWS_WG * 3; i += THREADS) Xs[i] = X[gRow0 * 3 + i];
  for (int i = tid; i < NC; i += THREADS) {
    Sacc[i] = 0.0f; Txa[i] = 0.0f; Tya[i] = 0.0f; Tza[i] = 0.0f;
  }
  __syncthreads();

  const int wave = tid >> 5, lane = tid & 31;
  const int half = lane >> 4, idx = lane & 15;
  const int rl = wave * 16;  // local row base of this wave
  const float sg = sigma[0];
  const float sig2 = sg * sg;

  // A operand: lanes 0-15 hold K0,K1 = (x,y); lanes 16-31 hold K2,K3 = (z,0)
  const int arow = rl + idx;
  v2f A;
  A.x = half ? Xs[arow * 3 + 2] : Xs[arow * 3 + 0];
  A.y = half ? 0.0f : Xs[arow * 3 + 1];

  const float4* muS = (const float4*)(ws + WS_MUS);
  const float* mu2t = ws + WS_MU2;

  // rows owned by this lane's 8 C/D slots: r_j = rl + j + 8*half
  float xr0[8], xr1[8], xr2[8], x2r[8];
#pragma unroll
  for (int j = 0; j < 8; ++j) {
    int r = rl + j + 8 * half;
    xr0[j] = Xs[r * 3 + 0]; xr1[j] = Xs[r * 3 + 1]; xr2[j] = Xs[r * 3 + 2];
    x2r[j] = xr0[j] * xr0[j] + xr1[j] * xr1[j] + xr2[j] * xr2[j];
  }

  // ---- sweep 1: online softmax over all 512 columns ----
  float mloc[8], sloc[8];
#pragma unroll
  for (int j = 0; j < 8; ++j) { mloc[j] = -3.0e38f; sloc[j] = 0.0f; }

  for (int t = 0; t < CTILES; ++t) {
    int c = t * 16 + idx;
    float4 rec = muS[c];
    v2f B; B.x = half ? rec.z : rec.x; B.y = half ? 0.0f : rec.y;
    v8f Cm;
#pragma unroll
    for (int j = 0; j < 8; ++j) Cm[j] = rec.w;  // bias b_c per column
    v8f Dv = wmma_score(A, B, Cm);
#pragma unroll
    for (int j = 0; j < 8; ++j) {
      float v = Dv[j];
      float nm = fmaxf(mloc[j], v);
      sloc[j] = sloc[j] * __expf(mloc[j] - nm) + __expf(v - nm);
      mloc[j] = nm;
    }
  }
  // merge (m,s) across the 16 lanes of each half (masks stay within half)
#pragma unroll
  for (int mask = 1; mask < 16; mask <<= 1) {
#pragma unroll
    for (int j = 0; j < 8; ++j) {
      float om = __shfl_xor(mloc[j], mask, 32);
      float os = __shfl_xor(sloc[j], mask, 32);
      float nm = fmaxf(mloc[j], om);
      sloc[j] = sloc[j] * __expf(mloc[j] - nm) + os * __expf(om - nm);
      mloc[j] = nm;
    }
  }
  float lse[8];
#pragma unroll
  for (int j = 0; j < 8; ++j) lse[j] = mloc[j] + __logf(sloc[j]);
  if (idx == 0) {
#pragma unroll
    for (int j = 0; j < 8; ++j) ws[WS_L + gRow0 + rl + j + 8 * half] = lse[j];
  }

  // ---- sweep 2: gamma, column sums, sum(gamma*D2) ----
  float gd2 = 0.0f;
  for (int t = 0; t < CTILES; ++t) {
    int c = t * 16 + idx;
    float4 rec = muS[c];
    float m2 = mu2t[c];
    v2f B; B.x = half ? rec.z : rec.x; B.y = half ? 0.0f : rec.y;
    v8f Cm;
#pragma unroll
    for (int j = 0; j < 8; ++j) Cm[j] = rec.w;
    v8f Dv = wmma_score(A, B, Cm);
    float cs = 0.0f, ctx = 0.0f, cty = 0.0f, ctz = 0.0f;
#pragma unroll
    for (int j = 0; j < 8; ++j) {
      float sc = Dv[j];
      float g = __expf(sc - lse[j]);
      cs += g; ctx += g * xr0[j]; cty += g * xr1[j]; ctz += g * xr2[j];
      // D2 = ||x||^2 + ||mu||^2 - 2*dot, dot = (score - b_c) * sigma^2
      float d2 = x2r[j] + m2 - 2.0f * sig2 * (sc - rec.w);
      gd2 += g * d2;
    }
    cs += __shfl_xor(cs, 16, 32);
    ctx += __shfl_xor(ctx, 16, 32);
    cty += __shfl_xor(cty, 16, 32);
    ctz += __shfl_xor(ctz, 16, 32);
    if (half == 0) {
      atomicAdd(&Sacc[c], cs);
      atomicAdd(&Txa[c], ctx);
      atomicAdd(&Tya[c], cty);
      atomicAdd(&Tza[c], ctz);
    }
  }
#pragma unroll
  for (int mask = 1; mask < 32; mask <<= 1) gd2 += __shfl_xor(gd2, mask, 32);
  if (lane == 0) atomicAdd(&ws[WS_GD2], gd2);

  __syncthreads();
  for (int i = tid; i < NC; i += THREADS) {
    atomicAdd(&ws[WS_S + i], Sacc[i]);
    atomicAdd(&ws[WS_TX + i], Txa[i]);
    atomicAdd(&ws[WS_TY + i], Tya[i]);
    atomicAdd(&ws[WS_TZ + i], Tza[i]);
  }
}

// ---------------- K2: M-step (single workgroup over C=512) ----------------
__global__ __launch_bounds__(512) void k2_mstep(float* __restrict__ ws,
                                                float* __restrict__ out) {
  __shared__ float red[512];
  const int c = threadIdx.x;
  float S = ws[WS_S + c];
  float invS = 1.0f / S;
  float mx = ws[WS_TX + c] * invS, my = ws[WS_TY + c] * invS,
        mz = ws[WS_TZ + c] * invS;
  float h = __logf(S);  // h_c = logsumexp_n lgamma = log(sum_n gamma)
  // logpi_c = w_new - lse(w_new) = h - logsumexp_c(h)  (mean term cancels)
  red[c] = h; __syncthreads();
  for (int s = 256; s > 0; s >>= 1) { if (c < s) red[c] = fmaxf(red[c], red[c + s]); __syncthreads(); }
  float hmax = red[0]; __syncthreads();
  red[c] = __expf(h - hmax); __syncthreads();
  for (int s = 256; s > 0; s >>= 1) { if (c < s) red[c] += red[c + s]; __syncthreads(); }
  float hlse = hmax + __logf(red[0]);
  float logpi = h - hlse;

  float sn2 = ws[WS_GD2] / (3.0f * (float)N_PTS);
  float sn = sqrtf(sn2);
  float inv2sn = 0.5f / sn2;
  float Ac = (mx * mx + my * my + mz * mz) * inv2sn - logpi;
  ((float4*)(ws + WS_REC2))[c] = make_float4(mx, my, mz, Ac);
  if (c == 0) {
    ws[WS_SCAL + 0] = inv2sn;
    ws[WS_SCAL + 1] = 3.0f * __logf(sn);
    out[3 * N_PTS] = 0.0f;  // Cfe accumulator
  }
}

// ---------------- K3: Y = gamma @ mu_new and Cfe ----------------
__global__ __launch_bounds__(THREADS) void k3_finalize(const float* __restrict__ X,
                                                       const float* __restrict__ ws,
                                                       float* __restrict__ out) {
  __shared__ float Xs[ROWS_WG * 3];
  const int tid = threadIdx.x;
  const int gRow0 = blockIdx.x * ROWS_WG;
  for (int i = tid; i < ROWS_WG * 3; i += THREADS) Xs[i] = X[gRow0 * 3 + i];
  __syncthreads();

  const int wave = tid >> 5, lane = tid & 31;
  const int half = lane >> 4, idx = lane & 15;
  const int rl = wave * 16;
  const int arow = rl + idx;
  v2f A;
  A.x = half ? Xs[arow * 3 + 2] : Xs[arow * 3 + 0];
  A.y = half ? 0.0f : Xs[arow * 3 + 1];

  const float4* muS = (const float4*)(ws + WS_MUS);
  const float4* rec2 = (const float4*)(ws + WS_REC2);
  const float inv2sn = ws[WS_SCAL + 0], dls = ws[WS_SCAL + 1];

  float lse[8];
#pragma unroll
  for (int j = 0; j < 8; ++j) lse[j] = ws[WS_L + gRow0 + rl + j + 8 * half];

  float Yx[8], Yy[8], Yz[8], ca[8];
#pragma unroll
  for (int j = 0; j < 8; ++j) { Yx[j] = 0; Yy[j] = 0; Yz[j] = 0; ca[j] = 0; }

  for (int t = 0; t < CTILES; ++t) {
    int c = t * 16 + idx;
    float4 rec = muS[c];
    v2f B; B.x = half ? rec.z : rec.x; B.y = half ? 0.0f : rec.y;
    v8f Cm;
#pragma unroll
    for (int j = 0; j < 8; ++j) Cm[j] = rec.w;
    v8f Dv = wmma_score(A, B, Cm);
    float4 r2 = rec2[c];  // (mu_new, A_c)
#pragma unroll
    for (int j = 0; j < 8; ++j) {
      float lg = Dv[j] - lse[j];  // lgamma
      float g = __expf(lg);
      Yx[j] += g * r2.x; Yy[j] += g * r2.y; Yz[j] += g * r2.z;
      ca[j] += g * (r2.w + lg);
    }
  }
  // reduce row sums across the 16 lanes of each half
#pragma unroll
  for (int mask = 1; mask < 16; mask <<= 1) {
#pragma unroll
    for (int j = 0; j < 8; ++j) {
      Yx[j] += __shfl_xor(Yx[j], mask, 32);
      Yy[j] += __shfl_xor(Yy[j], mask, 32);
      Yz[j] += __shfl_xor(Yz[j], mask, 32);
      ca[j] += __shfl_xor(ca[j], mask, 32);
    }
  }
  float cfe = 0.0f;
#pragma unroll
  for (int j = 0; j < 8; ++j) {
    if (idx == j) {  // lanes 0..7 write rows rl+0..7, lanes 16..23 rows rl+8..15
      int row = gRow0 + rl + j + 8 * half;
      out[row * 3 + 0] = Yx[j];
      out[row * 3 + 1] = Yy[j];
      out[row * 3 + 2] = Yz[j];
      float y2 = Yx[j] * Yx[j] + Yy[j] * Yy[j] + Yz[j] * Yz[j];
      cfe = ca[j] - y2 * inv2sn + dls;  // sum_c gamma = 1 folds -||Y||^2, d*log(sn)
    }
  }
#pragma unroll
  for (int mask = 1; mask < 32; mask <<= 1) cfe += __shfl_xor(cfe, mask, 32);
  if (lane == 0) atomicAdd(&out[3 * N_PTS], cfe);
}

extern "C" void kernel_launch(void* const* d_in, const int* in_sizes, int n_in,
                              void* d_out, int out_size, void* d_ws, size_t ws_size,
                              hipStream_t stream) {
  const float* X = (const float*)d_in[0];       // [N,3]
  const float* mu = (const float*)d_in[1];      // [C,3]
  const float* w = (const float*)d_in[2];       // [C]
  const float* sigma = (const float*)d_in[3];   // scalar
  float* out = (float*)d_out;                   // [N*3] Y then [1] Cfe
  float* ws = (float*)d_ws;                     // needs ~551KB

  k0_prep<<<1, 512, 0, stream>>>(mu, w, sigma, ws);
  k1_estep<<<N_PTS / ROWS_WG, THREADS, 0, stream>>>(X, sigma, ws);
  k2_mstep<<<1, 512, 0, stream>>>(ws, out);
  k3_finalize<<<N_PTS / ROWS_WG, THREADS, 0, stream>>>(X, ws, out);
}